// GHMC_88605175316656
// MI455X (gfx1250) — compile-verified
//
#include <hip/hip_runtime.h>
#include <hip/hip_bf16.h>

#define GHMC_BINS 10

// Bin index for an element; GHMC_BINS == "invalid" marker.
// Matches the reference's "last bin containing g" selection:
// bin = min(floor(g*10), 9) for g in [0,1).
__device__ __forceinline__ int ghmc_bin(float p, float l, float w) {
    if (!(w > 0.0f)) return GHMC_BINS;
    float sig = 1.0f / (1.0f + expf(-p));
    float g = fabsf(sig - l);
    int j = (int)(g * 10.0f);
    j = j > 9 ? 9 : (j < 0 ? 0 : j);
    return j;
}

__global__ void ghmc_zero_hist(unsigned int* __restrict__ hist) {
    if (threadIdx.x < 16) hist[threadIdx.x] = 0u;
}

__global__ __launch_bounds__(256) void ghmc_hist(
    const float4* __restrict__ pred, const float4* __restrict__ label,
    const float4* __restrict__ lw, unsigned int* __restrict__ hist,
    int n4, int n) {
    __shared__ unsigned int lhist[GHMC_BINS + 1];
    const int t = threadIdx.x;
    if (t < GHMC_BINS + 1) lhist[t] = 0u;
    __syncthreads();

    // wave-uniform counters (accumulated from 32-lane ballots)
    unsigned int cnt[GHMC_BINS + 1];
#pragma unroll
    for (int b = 0; b <= GHMC_BINS; ++b) cnt[b] = 0u;

    const int stride = gridDim.x * blockDim.x;
    for (int i = blockIdx.x * blockDim.x + t; i < n4; i += stride) {
        if (i + stride < n4) {
            __builtin_prefetch(&pred[i + stride], 0, 0);   // global_prefetch_b8
            __builtin_prefetch(&label[i + stride], 0, 0);
            __builtin_prefetch(&lw[i + stride], 0, 0);
        }
        float4 p = pred[i];
        float4 l = label[i];
        float4 w = lw[i];
        int j0 = ghmc_bin(p.x, l.x, w.x);
        int j1 = ghmc_bin(p.y, l.y, w.y);
        int j2 = ghmc_bin(p.z, l.z, w.z);
        int j3 = ghmc_bin(p.w, l.w, w.w);
#pragma unroll
        for (int b = 0; b < GHMC_BINS; ++b) {
            cnt[b] += (unsigned)__popcll(__ballot(j0 == b));
            cnt[b] += (unsigned)__popcll(__ballot(j1 == b));
            cnt[b] += (unsigned)__popcll(__ballot(j2 == b));
            cnt[b] += (unsigned)__popcll(__ballot(j3 == b));
        }
        cnt[GHMC_BINS] += (unsigned)__popcll(__ballot(j0 < GHMC_BINS));
        cnt[GHMC_BINS] += (unsigned)__popcll(__ballot(j1 < GHMC_BINS));
        cnt[GHMC_BINS] += (unsigned)__popcll(__ballot(j2 < GHMC_BINS));
        cnt[GHMC_BINS] += (unsigned)__popcll(__ballot(j3 < GHMC_BINS));
    }

    // one LDS atomic per wave per counter
    if ((t & (warpSize - 1)) == 0) {
#pragma unroll
        for (int b = 0; b <= GHMC_BINS; ++b)
            if (cnt[b]) atomicAdd(&lhist[b], cnt[b]);
    }
    __syncthreads();
    if (t < GHMC_BINS + 1 && lhist[t]) atomicAdd(&hist[t], lhist[t]);

    // tail elements (n not a multiple of 4) handled by one thread
    if (blockIdx.x == 0 && t == 0) {
        const float* ps = (const float*)pred;
        const float* ls = (const float*)label;
        const float* ws = (const float*)lw;
        for (int k = n4 * 4; k < n; ++k) {
            int j = ghmc_bin(ps[k], ls[k], ws[k]);
            if (j < GHMC_BINS) {
                atomicAdd(&hist[j], 1u);
                atomicAdd(&hist[GHMC_BINS], 1u);
            }
        }
    }
}

// Per-bin final scale: s_b = ((total_valid / max(n_b,1)) / valid_bins) / total_valid
// (reference's division ordering preserved; out = bce * s_b)
__global__ void ghmc_scale(const unsigned int* __restrict__ hist,
                           float* __restrict__ scale) {
    if (threadIdx.x == 0) {
        float tv = (float)hist[GHMC_BINS];
        tv = fmaxf(tv, 1.0f);
        float vb = 0.0f;
#pragma unroll
        for (int b = 0; b < GHMC_BINS; ++b) vb += (hist[b] > 0u) ? 1.0f : 0.0f;
#pragma unroll
        for (int b = 0; b < GHMC_BINS; ++b) {
            unsigned int nb = hist[b];
            float nf = fmaxf((float)nb, 1.0f);
            scale[b] = (nb > 0u) ? (((tv / nf) / vb) / tv) : 0.0f;
        }
    }
}

__device__ __forceinline__ float ghmc_elem(float p, float l, float w,
                                           const float* __restrict__ s) {
    if (!(w > 0.0f)) return 0.0f;
    float sig = 1.0f / (1.0f + expf(-p));
    float g = fabsf(sig - l);
    int j = (int)(g * 10.0f);
    j = j > 9 ? 9 : (j < 0 ? 0 : j);
    float logp   = fmaxf(logf(p), -100.0f);
    float log1mp = fmaxf(logf(1.0f - p), -100.0f);
    float bce = -(l * logp + (1.0f - l) * log1mp);
    return bce * s[j];
}

__global__ __launch_bounds__(256) void ghmc_out(
    const float4* __restrict__ pred, const float4* __restrict__ label,
    const float4* __restrict__ lw, const float* __restrict__ scaleG,
    float4* __restrict__ out, int n4, int n) {
    __shared__ float s[GHMC_BINS];
    if (threadIdx.x < GHMC_BINS) s[threadIdx.x] = scaleG[threadIdx.x];
    __syncthreads();

    int i = blockIdx.x * blockDim.x + threadIdx.x;
    if (i < n4) {
        float4 p = pred[i];
        float4 l = label[i];
        float4 w = lw[i];
        float4 o;
        o.x = ghmc_elem(p.x, l.x, w.x, s);
        o.y = ghmc_elem(p.y, l.y, w.y, s);
        o.z = ghmc_elem(p.z, l.z, w.z, s);
        o.w = ghmc_elem(p.w, l.w, w.w, s);
        out[i] = o;
    }

    if (blockIdx.x == 0 && threadIdx.x == 0) {
        const float* ps = (const float*)pred;
        const float* ls = (const float*)label;
        const float* ws = (const float*)lw;
        float* os = (float*)out;
        for (int k = n4 * 4; k < n; ++k)
            os[k] = ghmc_elem(ps[k], ls[k], ws[k], s);
    }
}

extern "C" void kernel_launch(void* const* d_in, const int* in_sizes, int n_in,
                              void* d_out, int out_size, void* d_ws, size_t ws_size,
                              hipStream_t stream) {
    const float* pred  = (const float*)d_in[0];
    const float* label = (const float*)d_in[1];
    const float* lwgt  = (const float*)d_in[2];
    float* out = (float*)d_out;

    const int n  = in_sizes[0];
    const int n4 = n >> 2;

    unsigned int* hist = (unsigned int*)d_ws;                // 11 u32 (+pad)
    float* scale = (float*)((char*)d_ws + 64);               // 10 f32

    ghmc_zero_hist<<<1, 32, 0, stream>>>(hist);

    {
        int blocks = 2048;
        int maxb = (n4 + 255) / 256;
        if (maxb < 1) maxb = 1;
        if (blocks > maxb) blocks = maxb;
        ghmc_hist<<<blocks, 256, 0, stream>>>(
            (const float4*)pred, (const float4*)label, (const float4*)lwgt,
            hist, n4, n);
    }

    ghmc_scale<<<1, 32, 0, stream>>>(hist, scale);

    {
        int blocks = (n4 + 255) / 256;
        if (blocks < 1) blocks = 1;
        ghmc_out<<<blocks, 256, 0, stream>>>(
            (const float4*)pred, (const float4*)label, (const float4*)lwgt,
            scale, (float4*)out, n4, n);
    }
}